// LMFSNN_71227737637176
// MI455X (gfx1250) — compile-verified
//
#include <hip/hip_runtime.h>
#include <math.h>

// LMFSNN forward for MI455X (gfx1250).
// Outputs (flat, in order): f [8192], u [8192], f_A [8192*8192].
// f_A write (256MB) is the bandwidth wall -> NT stores; per-point pinv
// replaced by normal equations G = A^T A solved with WMMA + Gauss-Jordan.

typedef float v2f __attribute__((ext_vector_type(2)));
typedef float v4f __attribute__((ext_vector_type(4)));
typedef float v8f __attribute__((ext_vector_type(8)));

#define NPTS 8192
#define NNL  30
#define NNS  15
#define WPB  8          // waves (points) per block

// ---------------------------------------------------------------- init ws
__global__ void lmfs_init_ws(unsigned int* __restrict__ ws) {
    if (threadIdx.x == 0 && blockIdx.x == 0) ws[0] = 0u;
}

// ------------------------------------------- zero fill of f_A + u copy
__global__ void __launch_bounds__(256)
lmfs_fill(const float* __restrict__ u, float* __restrict__ out) {
    float* fA = out + 2 * NPTS;
    const size_t total4 = (size_t)NPTS * (size_t)NPTS / 4;
    const size_t stride = (size_t)gridDim.x * blockDim.x;
    v4f z = {0.0f, 0.0f, 0.0f, 0.0f};
    for (size_t idx = (size_t)blockIdx.x * blockDim.x + threadIdx.x;
         idx < total4; idx += stride) {
        __builtin_nontemporal_store(z, ((v4f*)fA) + idx);
    }
    int tid = blockIdx.x * blockDim.x + threadIdx.x;
    if (tid < NPTS) out[NPTS + tid] = u[tid];   // second output: u.squeeze()
}

// ------------------------------------------------ global max ||x - tpx||^2
__global__ void __launch_bounds__(256)
lmfs_maxdm(const float* __restrict__ x, const float* __restrict__ centers,
           const int* __restrict__ nidx, unsigned int* __restrict__ ws) {
    __shared__ float red[256];
    const int total = NPTS * NNL;
    const int stride = gridDim.x * blockDim.x;
    float m = 0.0f;
    for (int t = blockIdx.x * blockDim.x + threadIdx.x; t < total; t += stride) {
        int i  = t / NNL;
        int nb = nidx[t];
        float dx = x[2 * i]     - centers[2 * nb];
        float dy = x[2 * i + 1] - centers[2 * nb + 1];
        m = fmaxf(m, dx * dx + dy * dy);
    }
    red[threadIdx.x] = m;
    __syncthreads();
    for (int s = 128; s > 0; s >>= 1) {
        if (threadIdx.x < (unsigned)s)
            red[threadIdx.x] = fmaxf(red[threadIdx.x], red[threadIdx.x + s]);
        __syncthreads();
    }
    // d2 >= 0 -> float bits are order-preserving as uint
    if (threadIdx.x == 0) atomicMax(ws, __float_as_uint(red[0]));
}

// --------------------------------------------------------- main per-point
// One wave per point. A[32][16] in LDS (rows>=30 and col 15 zero-padded).
// G = A^T A via 8x V_WMMA_F32_16X16X4_F32; solve G y = mfs with cooperative
// Gauss-Jordan (G is SPD); w = A y; scatter -w into f_A; f = u_i - sum w*u_nb.
__global__ void __launch_bounds__(WPB * 32)
lmfs_main(const float* __restrict__ x, const float* __restrict__ centers,
          const float* __restrict__ u, const float* __restrict__ R,
          const int* __restrict__ nidx, const unsigned int* __restrict__ ws,
          float* __restrict__ out) {
    __shared__ float A_s[WPB][32][16];   // mfs1 (log distances), padded
    __shared__ float M_s[WPB][16][17];   // [G | b] augmented (b lives in col 15)
    __shared__ float c_s[WPB][16][2];    // circle sample points
    __shared__ float red_s[WPB][32];

    const int w    = threadIdx.x >> 5;
    const int lane = threadIdx.x & 31;
    const int i    = blockIdx.x * WPB + w;

    const float maxdm = sqrtf(__uint_as_float(ws[0]));

    // zero the padded A tile
    for (int t = lane; t < 512; t += 32) ((float*)A_s[w])[t] = 0.0f;

    const float xi  = x[2 * i],       yi  = x[2 * i + 1];
    const float cxi = centers[2 * i], cyi = centers[2 * i + 1];
    const float rad = R[i] * maxdm;

    float bval = 0.0f;
    if (lane < NNS) {
        float ang = (6.28318530717958647692f * (float)lane) / (float)NNS;
        float cx = cxi + rad * cosf(ang);
        float cy = cyi + rad * sinf(ang);
        c_s[w][lane][0] = cx;
        c_s[w][lane][1] = cy;
        float dx = xi - cx, dy = yi - cy;
        bval = 0.5f * __logf(dx * dx + dy * dy);      // mfs[s] = log(dm)
    }

    int   nb = 0;
    float tx = 0.0f, ty = 0.0f;
    if (lane < NNL) {
        nb = nidx[i * NNL + lane];
        tx = centers[2 * nb];
        ty = centers[2 * nb + 1];
    }
    __builtin_amdgcn_wave_barrier();    // order LDS writes of c_s before reads

    if (lane < NNL) {
#pragma unroll
        for (int s = 0; s < NNS; ++s) {
            float dx = tx - c_s[w][s][0];
            float dy = ty - c_s[w][s][1];
            A_s[w][lane][s] = 0.5f * __logf(dx * dx + dy * dy);  // mfs1[l][s]
        }
    }
    __builtin_amdgcn_wave_barrier();

    // ---- G = A^T A  (16x16 accum over K=32) via WMMA f32 16x16x4 ----
    // A-operand tile (A^T, 16x4) and B-operand tile (A, 4x16) read the same
    // LDS values because G is symmetric and the operand layouts mirror.
    const int m  = lane & 15;
    const int hi = lane >> 4;
    v8f acc = {0.0f, 0.0f, 0.0f, 0.0f, 0.0f, 0.0f, 0.0f, 0.0f};
#pragma unroll
    for (int kb = 0; kb < 8; ++kb) {
        const int k0 = kb * 4 + hi * 2;
        v2f t;
        t.x = A_s[w][k0][m];        // A^T[m][k0]   == A[k0][m]
        t.y = A_s[w][k0 + 1][m];    // A^T[m][k0+1] == A[k0+1][m]
        acc = __builtin_amdgcn_wmma_f32_16x16x4_f32(
            /*neg_a=*/false, t, /*neg_b=*/false, t,
            /*c_mod=*/(short)0, acc, /*reuse_a=*/false, /*reuse_b=*/false);
    }

    // spill G into LDS per documented C/D layout: VGPR r -> row r + 8*hi
#pragma unroll
    for (int r = 0; r < 8; ++r) M_s[w][r + hi * 8][m] = acc[r];

    // RHS b into column 15 (G's column 15 is structurally zero: A col 15 = 0)
    if (lane < NNS) M_s[w][lane][15] = bval;
    __builtin_amdgcn_wave_barrier();

    // ---- cooperative Gauss-Jordan on [G | b], 15x15 SPD, lanes = columns ----
#pragma unroll 1
    for (int p = 0; p < NNS; ++p) {
        float inv = 1.0f / M_s[w][p][p];
        float rowpj = 0.0f;
        if (lane < 16) {
            rowpj = M_s[w][p][lane] * inv;
            M_s[w][p][lane] = rowpj;
        }
#pragma unroll 1
        for (int r = 0; r < NNS; ++r) {
            if (r == p) continue;
            float fac = M_s[w][r][p];           // read (lockstep) before write
            if (lane < 16)
                M_s[w][r][lane] = fmaf(-fac, rowpj, M_s[w][r][lane]);
        }
    }
    __builtin_amdgcn_wave_barrier();

    // ---- w_l = sum_s A[l][s] * y[s];  y[s] = M_s[s][15] ----
    float partial = 0.0f;
    float* fA = out + 2 * NPTS;
    if (lane < NNL) {
        float wl = 0.0f;
#pragma unroll
        for (int s = 0; s < NNS; ++s)
            wl = fmaf(A_s[w][lane][s], M_s[w][s][15], wl);
        fA[(size_t)i * NPTS + nb] = -wl;        // scatter into f_A row i
        partial = wl * u[nb];
    }
    if (lane == 31) fA[(size_t)i * NPTS + i] = 1.0f;   // identity diagonal

    red_s[w][lane] = partial;
    __builtin_amdgcn_wave_barrier();
    if (lane == 0) {
        float ssum = 0.0f;
#pragma unroll
        for (int t2 = 0; t2 < 32; ++t2) ssum += red_s[w][t2];
        out[i] = u[i] - ssum;                   // f = f_A @ u, row i
    }
}

// ----------------------------------------------------------------- launch
extern "C" void kernel_launch(void* const* d_in, const int* in_sizes, int n_in,
                              void* d_out, int out_size, void* d_ws, size_t ws_size,
                              hipStream_t stream) {
    (void)in_sizes; (void)n_in; (void)out_size; (void)ws_size;
    const float* x       = (const float*)d_in[0];
    const float* centers = (const float*)d_in[1];
    const float* u       = (const float*)d_in[2];
    const float* R       = (const float*)d_in[3];
    const int*   nidx    = (const int*)d_in[4];
    float*       out     = (float*)d_out;
    unsigned int* ws     = (unsigned int*)d_ws;

    lmfs_init_ws<<<1, 32, 0, stream>>>(ws);
    lmfs_fill<<<4096, 256, 0, stream>>>(u, out);                 // 256MB NT fill
    lmfs_maxdm<<<240, 256, 0, stream>>>(x, centers, nidx, ws);
    lmfs_main<<<NPTS / WPB, WPB * 32, 0, stream>>>(x, centers, u, R, nidx, ws, out);
}